// GNNEncoder_15659450761724
// MI455X (gfx1250) — compile-verified
//
#include <hip/hip_runtime.h>
#include <hip/hip_bf16.h>
#include <math.h>

#define N_NODES 50000
#define N_EDGES 800000
#define DH      64
#define NHEAD   4
#define HD      256   // NHEAD * DH

typedef float v2f __attribute__((ext_vector_type(2)));
typedef float v8f __attribute__((ext_vector_type(8)));

// float atomic max via sign-split integer atomics (works for all non-NaN floats)
__device__ __forceinline__ void atomic_max_f32(float* addr, float val) {
  if (val >= 0.0f) atomicMax((int*)addr, __float_as_int(val));
  else             atomicMin((unsigned int*)addr, __float_as_uint(val));
}

// -------------------------------------------------------------------------
// init: fill attention-max buffer with -inf
// -------------------------------------------------------------------------
__global__ void fill_neg_inf(float* __restrict__ m, int n) {
  int idx = blockIdx.x * blockDim.x + threadIdx.x;
  if (idx < n) m[idx] = __int_as_float(0xff800000u); // -inf
}

// -------------------------------------------------------------------------
// K2: h = emb[text_idx];  fs = h@W_src+b ; fd = h@W_dst+b ; res = h@W_res
// 16 rows per block, 8 waves, each wave computes 6 of 48 16x16 output tiles.
// K=64 via 16 chained V_WMMA_F32_16X16X4_F32.
// A layout (16x4 f32): element(m,k) -> lane = m + 16*(k>>1), vgpr = k&1
// B layout (4x16 f32): element(k,n) -> lane = n + 16*(k>>1), vgpr = k&1
// C/D (16x16 f32):     element(m,n) -> vgpr = m&7, lane = n + 16*(m>>3)
// -------------------------------------------------------------------------
__global__ __launch_bounds__(256) void node_transform_wmma(
    const int*   __restrict__ text_idx, const float* __restrict__ emb,
    const float* __restrict__ W_src,    const float* __restrict__ b_src,
    const float* __restrict__ W_dst,    const float* __restrict__ b_dst,
    const float* __restrict__ W_res,
    float* __restrict__ fs, float* __restrict__ fd, float* __restrict__ res)
{
  __shared__ float sh[16][65];            // padded stride: no bank conflicts
  const int row0 = blockIdx.x * 16;       // N divisible by 16
  {
    const int idx = threadIdx.x * 4;      // 1024 elems / 256 threads
    const int r = idx >> 6, c = idx & 63; // same row for all 4 elems
    const int tid = text_idx[row0 + r];
    const float4 v = *(const float4*)(emb + tid * 64 + c);
    sh[r][c] = v.x; sh[r][c + 1] = v.y; sh[r][c + 2] = v.z; sh[r][c + 3] = v.w;
  }
  __syncthreads();

  const int lane = threadIdx.x & 31;
  const int wave = threadIdx.x >> 5;
  const int nl   = lane & 15;             // n within tile (also m for A rows)
  const int grp  = lane >> 4;             // half-wave group

  for (int t = wave; t < 48; t += 8) {    // tile id: [mat 0..2][ntile 0..15]
    const int mat = t >> 4;
    const int nt  = t & 15;
    const float* W = (mat == 0) ? W_src : (mat == 1) ? W_dst : W_res;
    const int ncol = nt * 16 + nl;
    v8f acc = {0.f, 0.f, 0.f, 0.f, 0.f, 0.f, 0.f, 0.f};
#pragma unroll
    for (int kb = 0; kb < 16; ++kb) {
      const int k0 = kb * 4 + grp * 2;
      v2f a, b;
      a.x = sh[nl][k0];
      a.y = sh[nl][k0 + 1];
      b.x = W[k0 * HD + ncol];
      b.y = W[(k0 + 1) * HD + ncol];
      acc = __builtin_amdgcn_wmma_f32_16x16x4_f32(
          /*neg_a=*/false, a, /*neg_b=*/false, b,
          /*c_mod=*/(short)0, acc, /*reuse_a=*/false, /*reuse_b=*/false);
    }
    const float bias = (mat == 0) ? b_src[ncol] : (mat == 1) ? b_dst[ncol] : 0.0f;
    float* out = (mat == 0) ? fs : (mat == 1) ? fd : res;
#pragma unroll
    for (int r = 0; r < 8; ++r) {
      const int m = r + grp * 8;
      out[(row0 + m) * HD + ncol] = acc[r] + bias;
    }
  }
}

// -------------------------------------------------------------------------
// K3: per-edge attention logits + segment max + degree counts.
// One wave per edge; lane L handles 8 contiguous floats; 8 lanes per head.
// -------------------------------------------------------------------------
__global__ __launch_bounds__(256) void edge_logits(
    const int* __restrict__ src, const int* __restrict__ dst,
    const float* __restrict__ fs, const float* __restrict__ fd,
    const float* __restrict__ attn,
    float* __restrict__ logits, float* __restrict__ mx,
    float* __restrict__ outdeg, float* __restrict__ indeg)
{
  const int lane = threadIdx.x & 31;
  const int e = blockIdx.x * 8 + (threadIdx.x >> 5);
  if (e >= N_EDGES) return;
  const int s = src[e], d = dst[e];
  const int base = lane * 8;
  const float4 s0 = *(const float4*)(fs + s * HD + base);
  const float4 s1 = *(const float4*)(fs + s * HD + base + 4);
  const float4 d0 = *(const float4*)(fd + d * HD + base);
  const float4 d1 = *(const float4*)(fd + d * HD + base + 4);
  const float4 a0 = *(const float4*)(attn + base);
  const float4 a1 = *(const float4*)(attn + base + 4);
  float sum = 0.f, x;
  x = s0.x + d0.x; sum += (x > 0.f ? x : 0.2f * x) * a0.x;
  x = s0.y + d0.y; sum += (x > 0.f ? x : 0.2f * x) * a0.y;
  x = s0.z + d0.z; sum += (x > 0.f ? x : 0.2f * x) * a0.z;
  x = s0.w + d0.w; sum += (x > 0.f ? x : 0.2f * x) * a0.w;
  x = s1.x + d1.x; sum += (x > 0.f ? x : 0.2f * x) * a1.x;
  x = s1.y + d1.y; sum += (x > 0.f ? x : 0.2f * x) * a1.y;
  x = s1.z + d1.z; sum += (x > 0.f ? x : 0.2f * x) * a1.z;
  x = s1.w + d1.w; sum += (x > 0.f ? x : 0.2f * x) * a1.w;
  // reduce within 8-lane head groups (wave32)
  sum += __shfl_xor(sum, 1, 32);
  sum += __shfl_xor(sum, 2, 32);
  sum += __shfl_xor(sum, 4, 32);
  const int head = lane >> 3;
  if ((lane & 7) == 0) {
    logits[e * NHEAD + head] = sum;
    atomic_max_f32(&mx[d * NHEAD + head], sum);
  }
  if (lane == 0) {
    unsafeAtomicAdd(&outdeg[s], 1.0f);
    unsafeAtomicAdd(&indeg[d], 1.0f);
  }
}

// -------------------------------------------------------------------------
// K4: p = exp(logit - max[dst]); segment sum of p.  One thread per (e,h).
// -------------------------------------------------------------------------
__global__ void edge_softmax_sum(const int* __restrict__ dst,
                                 const float* __restrict__ logits,
                                 const float* __restrict__ mx,
                                 float* __restrict__ p,
                                 float* __restrict__ ssum)
{
  const int idx = blockIdx.x * blockDim.x + threadIdx.x;
  if (idx >= N_EDGES * NHEAD) return;
  const int e = idx >> 2, h = idx & 3;
  const int d = dst[e];
  const float v = __expf(logits[idx] - mx[d * NHEAD + h]);
  p[idx] = v;
  unsafeAtomicAdd(&ssum[d * NHEAD + h], v);
}

// -------------------------------------------------------------------------
// K5: rst[dst] += (p/ssum[dst]) * fs[src].  One wave per edge.
// -------------------------------------------------------------------------
__global__ __launch_bounds__(256) void edge_aggregate(
    const int* __restrict__ src, const int* __restrict__ dst,
    const float* __restrict__ fs, const float* __restrict__ p,
    const float* __restrict__ ssum, float* __restrict__ rst)
{
  const int lane = threadIdx.x & 31;
  const int e = blockIdx.x * 8 + (threadIdx.x >> 5);
  if (e >= N_EDGES) return;
  const int s = src[e], d = dst[e];
  const int head = lane >> 3;
  const float a = p[e * NHEAD + head] / ssum[d * NHEAD + head];
  const int base = lane * 8;
  const float4 f0 = *(const float4*)(fs + s * HD + base);
  const float4 f1 = *(const float4*)(fs + s * HD + base + 4);
  float* out = rst + d * HD + base;
  unsafeAtomicAdd(out + 0, a * f0.x);
  unsafeAtomicAdd(out + 1, a * f0.y);
  unsafeAtomicAdd(out + 2, a * f0.z);
  unsafeAtomicAdd(out + 3, a * f0.w);
  unsafeAtomicAdd(out + 4, a * f1.x);
  unsafeAtomicAdd(out + 5, a * f1.y);
  unsafeAtomicAdd(out + 6, a * f1.z);
  unsafeAtomicAdd(out + 7, a * f1.w);
}

// -------------------------------------------------------------------------
// K6: h1 = relu(rst + res).mean(heads) * outdeg^-0.5  -> h1s [N,64]
// -------------------------------------------------------------------------
__global__ void node_mean_relu(const float* __restrict__ rst,
                               const float* __restrict__ res,
                               const float* __restrict__ outdeg,
                               float* __restrict__ h1s)
{
  const int idx = blockIdx.x * blockDim.x + threadIdx.x;
  if (idx >= N_NODES * DH) return;
  const int n = idx >> 6, dd = idx & 63;
  float acc = 0.f;
#pragma unroll
  for (int h = 0; h < NHEAD; ++h) {
    const float v = rst[n * HD + h * DH + dd] + res[n * HD + h * DH + dd];
    acc += (v > 0.f ? v : 0.f);
  }
  acc *= 0.25f;
  float od = outdeg[n]; od = (od < 1.f) ? 1.f : od;
  h1s[idx] = acc * rsqrtf(od);
}

// -------------------------------------------------------------------------
// K7: hh = h1s @ Wg (64x64) via WMMA.  32 rows per block, 8 waves = 8 tiles.
// -------------------------------------------------------------------------
__global__ __launch_bounds__(256) void gconv_wmma(
    const float* __restrict__ h1s, const float* __restrict__ Wg,
    float* __restrict__ hh)
{
  __shared__ float sh[32][65];
  const int row0 = blockIdx.x * 32;
  {
    const int idx = threadIdx.x * 8;      // 2048 elems / 256 threads
#pragma unroll
    for (int j = 0; j < 2; ++j) {
      const int id2 = idx + j * 4;
      const int r = id2 >> 6, c = id2 & 63;
      const int row = row0 + r;
      float4 v = make_float4(0.f, 0.f, 0.f, 0.f);
      if (row < N_NODES) v = *(const float4*)(h1s + row * 64 + c);
      sh[r][c] = v.x; sh[r][c + 1] = v.y; sh[r][c + 2] = v.z; sh[r][c + 3] = v.w;
    }
  }
  __syncthreads();

  const int lane = threadIdx.x & 31;
  const int wave = threadIdx.x >> 5;
  const int nl = lane & 15, grp = lane >> 4;
  const int rbase = (wave >> 2) * 16;     // row group 0/1
  const int ncol  = (wave & 3) * 16 + nl; // 4 col tiles
  v8f acc = {0.f, 0.f, 0.f, 0.f, 0.f, 0.f, 0.f, 0.f};
#pragma unroll
  for (int kb = 0; kb < 16; ++kb) {
    const int k0 = kb * 4 + grp * 2;
    v2f a, b;
    a.x = sh[rbase + nl][k0];
    a.y = sh[rbase + nl][k0 + 1];
    b.x = Wg[k0 * 64 + ncol];
    b.y = Wg[(k0 + 1) * 64 + ncol];
    acc = __builtin_amdgcn_wmma_f32_16x16x4_f32(false, a, false, b,
                                                (short)0, acc, false, false);
  }
#pragma unroll
  for (int r = 0; r < 8; ++r) {
    const int row = row0 + rbase + r + grp * 8;
    if (row < N_NODES) hh[row * 64 + ncol] = acc[r];
  }
}

// -------------------------------------------------------------------------
// K8: agg[dst] += hh[src].  One wave per edge, float2 per lane.
// -------------------------------------------------------------------------
__global__ __launch_bounds__(256) void edge_gconv_agg(
    const int* __restrict__ src, const int* __restrict__ dst,
    const float* __restrict__ hh, float* __restrict__ agg)
{
  const int lane = threadIdx.x & 31;
  const int e = blockIdx.x * 8 + (threadIdx.x >> 5);
  if (e >= N_EDGES) return;
  const int s = src[e], d = dst[e];
  const float2 v = *(const float2*)(hh + s * 64 + lane * 2);
  unsafeAtomicAdd(&agg[d * 64 + lane * 2 + 0], v.x);
  unsafeAtomicAdd(&agg[d * 64 + lane * 2 + 1], v.y);
}

// -------------------------------------------------------------------------
// K9: out = relu(agg * indeg^-0.5 + bg)
// -------------------------------------------------------------------------
__global__ void node_out(const float* __restrict__ agg,
                         const float* __restrict__ indeg,
                         const float* __restrict__ bg,
                         float* __restrict__ out)
{
  const int idx = blockIdx.x * blockDim.x + threadIdx.x;
  if (idx >= N_NODES * DH) return;
  const int n = idx >> 6, dd = idx & 63;
  float id = indeg[n]; id = (id < 1.f) ? 1.f : id;
  const float v = agg[idx] * rsqrtf(id) + bg[dd];
  out[idx] = (v > 0.f ? v : 0.f);
}

// -------------------------------------------------------------------------
extern "C" void kernel_launch(void* const* d_in, const int* in_sizes, int n_in,
                              void* d_out, int out_size, void* d_ws, size_t ws_size,
                              hipStream_t stream) {
  const int*   text_idx = (const int*)  d_in[0];
  const int*   src      = (const int*)  d_in[1];
  const int*   dst      = (const int*)  d_in[2];
  const float* emb      = (const float*)d_in[3];
  const float* W_src    = (const float*)d_in[4];
  const float* b_src    = (const float*)d_in[5];
  const float* W_dst    = (const float*)d_in[6];
  const float* b_dst    = (const float*)d_in[7];
  const float* attn     = (const float*)d_in[8];
  const float* W_res    = (const float*)d_in[9];
  const float* Wg       = (const float*)d_in[10];
  const float* bg       = (const float*)d_in[11];
  float* out = (float*)d_out;

  float* ws = (float*)d_ws;
  size_t off = 0;
  float* fs     = ws + off; off += (size_t)N_NODES * HD;
  float* fd     = ws + off; off += (size_t)N_NODES * HD;
  float* res    = ws + off; off += (size_t)N_NODES * HD;
  float* rst    = ws + off; off += (size_t)N_NODES * HD;
  float* logits = ws + off; off += (size_t)N_EDGES * NHEAD;
  float* p      = ws + off; off += (size_t)N_EDGES * NHEAD;
  float* mx     = ws + off; off += (size_t)N_NODES * NHEAD;
  float* ssum   = ws + off; off += (size_t)N_NODES * NHEAD;
  float* outdeg = ws + off; off += (size_t)N_NODES;
  float* indeg  = ws + off; off += (size_t)N_NODES;
  float* h1s    = ws + off; off += (size_t)N_NODES * DH;
  float* hh     = ws + off; off += (size_t)N_NODES * DH;
  float* agg    = ws + off; off += (size_t)N_NODES * DH;

  hipMemsetAsync(rst,    0, (size_t)N_NODES * HD * sizeof(float), stream);
  hipMemsetAsync(ssum,   0, (size_t)N_NODES * NHEAD * sizeof(float), stream);
  hipMemsetAsync(outdeg, 0, (size_t)N_NODES * sizeof(float), stream);
  hipMemsetAsync(indeg,  0, (size_t)N_NODES * sizeof(float), stream);
  hipMemsetAsync(agg,    0, (size_t)N_NODES * DH * sizeof(float), stream);
  fill_neg_inf<<<(N_NODES * NHEAD + 255) / 256, 256, 0, stream>>>(mx, N_NODES * NHEAD);

  node_transform_wmma<<<N_NODES / 16, 256, 0, stream>>>(
      text_idx, emb, W_src, b_src, W_dst, b_dst, W_res, fs, fd, res);
  edge_logits<<<N_EDGES / 8, 256, 0, stream>>>(
      src, dst, fs, fd, attn, logits, mx, outdeg, indeg);
  edge_softmax_sum<<<(N_EDGES * NHEAD + 255) / 256, 256, 0, stream>>>(
      dst, logits, mx, p, ssum);
  edge_aggregate<<<N_EDGES / 8, 256, 0, stream>>>(
      src, dst, fs, p, ssum, rst);
  node_mean_relu<<<(N_NODES * DH + 255) / 256, 256, 0, stream>>>(
      rst, res, outdeg, h1s);
  gconv_wmma<<<(N_NODES + 31) / 32, 256, 0, stream>>>(h1s, Wg, hh);
  edge_gconv_agg<<<N_EDGES / 8, 256, 0, stream>>>(src, dst, hh, agg);
  node_out<<<(N_NODES * DH + 255) / 256, 256, 0, stream>>>(agg, indeg, bg, out);
}